// LinearAttention_52690658788151
// MI455X (gfx1250) — compile-verified
//
#include <hip/hip_runtime.h>

typedef __attribute__((ext_vector_type(2))) float v2f;
typedef __attribute__((ext_vector_type(8))) float v8f;

#define NTOK 262144   // 64*64*64 voxels per batch
#define CCH  64       // channels
#define HIDDEN 128

#define G_BLOCKS_PER_BATCH 64   // gram: 4096 tokens/block, 1024/wave
#define Y_BLOCKS_PER_BATCH 256  // main: 1024 tokens/block, 256/wave

__device__ __forceinline__ v8f v8f_zero() {
    v8f z;
#pragma unroll
    for (int i = 0; i < 8; ++i) z[i] = 0.0f;
    return z;
}

// D = A(16x4 f32) * B(4x16 f32) + C(16x16 f32)
__device__ __forceinline__ v8f wmma_f32(v2f a, v2f b, v8f c) {
    // 8 args: (neg_a, A, neg_b, B, c_mod, C, reuse_a, reuse_b)
    return __builtin_amdgcn_wmma_f32_16x16x4_f32(false, a, false, b, (short)0, c,
                                                 false, false);
}

// ---------------------------------------------------------------- kernel 0
__global__ void zero_kernel(float* __restrict__ p, int count) {
    int i = blockIdx.x * blockDim.x + threadIdx.x;
    if (i < count) p[i] = 0.0f;
}

// ---------------------------------------------------------------- kernel 1
// Per-batch Gram matrix G = X * X^T  (64x64), accumulated via f32 WMMA.
// Each wave: stage 64ch x 32tok into its private LDS tile (coalesced loads),
// read back channel-major (pitch 34 => conflict-free, 8B aligned for v2f),
// run 16 WMMAs per 4-token k-group. By symmetry the A and B operands for
// channel-groups are the SAME registers (X vs X^T), so 4 v2f loads feed all 16.
__global__ __launch_bounds__(128) void gram_kernel(const float* __restrict__ x,
                                                   float* __restrict__ G) {
    const int tid    = threadIdx.x;
    const int lane   = tid & 31;
    const int w      = tid >> 5;
    const int batch  = blockIdx.x / G_BLOCKS_PER_BATCH;
    const int blk    = blockIdx.x % G_BLOCKS_PER_BATCH;
    const float* xb  = x + (size_t)batch * CCH * NTOK;
    float* Gb        = G + batch * CCH * CCH;

    __shared__ float tile[4][CCH][34];   // pitch 34: even (v2f-aligned) + conflict-free

    const int lanelo = lane & 15;
    const int hi2    = (lane >> 4) * 2;      // lanes 16-31 hold K=2,3
    const int rowOff = (lane < 16) ? 0 : 8;  // C/D layout M offset

    v8f acc[4][4];
#pragma unroll
    for (int i = 0; i < 4; ++i)
#pragma unroll
        for (int j = 0; j < 4; ++j) acc[i][j] = v8f_zero();

    const int tokensPerBlock = NTOK / G_BLOCKS_PER_BATCH;  // 4096
    const int tokensPerWave  = tokensPerBlock / 4;         // 1024
    const int tWave0         = blk * tokensPerBlock + w * tokensPerWave;

    for (int it = 0; it < tokensPerWave; it += 32) {
        const int t0 = tWave0 + it;
        // stage: lane = token -> fully coalesced 128B per instruction
#pragma unroll
        for (int c = 0; c < CCH; ++c)
            tile[w][c][lane] = xb[(size_t)c * NTOK + t0 + lane];
        __syncthreads();  // uniform across all 4 waves

#pragma unroll
        for (int kg = 0; kg < 8; ++kg) {   // 8 k-groups of 4 tokens
            v2f op[4];
#pragma unroll
            for (int gch = 0; gch < 4; ++gch) {
                const float* p = &tile[w][gch * 16 + lanelo][kg * 4 + hi2];
                op[gch] = *(const v2f*)p;  // (ch*34 + even) even -> 8B aligned
            }
#pragma unroll
            for (int i = 0; i < 4; ++i)
#pragma unroll
                for (int j = 0; j < 4; ++j)
                    acc[i][j] = wmma_f32(op[i], op[j], acc[i][j]);
        }
        __syncthreads();
    }

    // spill partial Gram: C/D layout -> G[i*16 + r + rowOff][j*16 + lanelo]
#pragma unroll
    for (int i = 0; i < 4; ++i)
#pragma unroll
        for (int j = 0; j < 4; ++j)
#pragma unroll
            for (int r = 0; r < 8; ++r) {
                const int row = i * 16 + r + rowOff;
                const int col = j * 16 + lanelo;
                atomicAdd(&Gb[row * CCH + col], acc[i][j][r]);
            }
}

// ---------------------------------------------------------------- kernel 2
// Collapse:  A_b = Wout * M_b,  M_b[h*32+e][c] = scale * sum_d ctx[h][d][e]*Wq[h*32+d][c],
//            ctx = (Wk * G * Wv^T)/4096.  One block per batch; tiny work.
__global__ __launch_bounds__(256) void combine_kernel(
    const float* __restrict__ wqkv, const float* __restrict__ wout,
    const float* __restrict__ G, float* __restrict__ T1, float* __restrict__ CT,
    float* __restrict__ M, float* __restrict__ A) {
    const int b   = blockIdx.x;
    const int tid = threadIdx.x;
    const float* Gb  = G  + b * 4096;
    float* T1b       = T1 + b * 8192;   // [64][128]
    float* CTb       = CT + b * 4096;   // [4][32][32]
    float* Mb        = M  + b * 8192;   // [128][64]
    float* Ab        = A  + b * 4096;   // [64][64]
    const float* Wq  = wqkv;                 // rows   0..127
    const float* Wk  = wqkv + 128 * CCH;     // rows 128..255
    const float* Wv  = wqkv + 256 * CCH;     // rows 256..383
    const float scale = 0.1767766952966369f; // 32^-0.5

    // T1[c][j] = sum_c2 G[c][c2] * Wv[j][c2]
    for (int idx = tid; idx < 64 * 128; idx += 256) {
        const int c = idx >> 7, j = idx & 127;
        float s = 0.f;
        for (int c2 = 0; c2 < CCH; ++c2) s += Gb[c * CCH + c2] * Wv[j * CCH + c2];
        T1b[idx] = s;
    }
    __syncthreads();
    // ctx[h][d][e] = (1/4096) sum_c Wk[h*32+d][c] * T1[c][h*32+e]
    for (int idx = tid; idx < 4096; idx += 256) {
        const int h = idx >> 10, d = (idx >> 5) & 31, e = idx & 31;
        float s = 0.f;
        for (int c = 0; c < CCH; ++c)
            s += Wk[(h * 32 + d) * CCH + c] * T1b[c * 128 + h * 32 + e];
        CTb[idx] = s * (1.0f / 4096.0f);
    }
    __syncthreads();
    // M[j][c] = scale * sum_d ctx[h][d][e] * Wq[h*32+d][c]   (j = h*32+e)
    for (int idx = tid; idx < 128 * 64; idx += 256) {
        const int j = idx >> 6, c = idx & 63, h = j >> 5, e = j & 31;
        float s = 0.f;
        for (int d = 0; d < 32; ++d)
            s += CTb[h * 1024 + d * 32 + e] * Wq[(h * 32 + d) * CCH + c];
        Mb[idx] = s * scale;
    }
    __syncthreads();
    // A[o][c] = sum_j Wout[o][j] * M[j][c]
    for (int idx = tid; idx < 4096; idx += 256) {
        const int o = idx >> 6, c = idx & 63;
        float s = 0.f;
        for (int j = 0; j < HIDDEN; ++j) s += wout[o * HIDDEN + j] * Mb[j * 64 + c];
        Ab[idx] = s;
    }
}

// ---------------------------------------------------------------- kernel 3
// y = A_b * x + b_out, then channel LayerNorm * g.  WMMA: M=out-ch, N=token,
// K=in-ch.  B-operand x loads are fully coalesced (token-contiguous memory).
__global__ __launch_bounds__(128) void attn_out_kernel(
    const float* __restrict__ x, const float* __restrict__ A,
    const float* __restrict__ b_out, const float* __restrict__ g,
    float* __restrict__ out) {
    const int tid    = threadIdx.x;
    const int lane   = tid & 31;
    const int w      = tid >> 5;
    const int batch  = blockIdx.x / Y_BLOCKS_PER_BATCH;
    const int blk    = blockIdx.x % Y_BLOCKS_PER_BATCH;
    const float* xb  = x + (size_t)batch * CCH * NTOK;
    float* ob        = out + (size_t)batch * CCH * NTOK;
    const float* Ab  = A + batch * CCH * CCH;

    const int lanelo = lane & 15;
    const int hi2    = (lane >> 4) * 2;
    const int rowOff = (lane < 16) ? 0 : 8;

    // Preload all A operands once per wave: aop[kg][mt] (16 KB matrix, L2-hot)
    v2f aop[16][4];
#pragma unroll
    for (int kg = 0; kg < 16; ++kg)
#pragma unroll
        for (int mt = 0; mt < 4; ++mt) {
            const float* p = &Ab[(mt * 16 + lanelo) * CCH + kg * 4 + hi2];
            aop[kg][mt] = *(const v2f*)p;  // consecutive K -> 8B-aligned v2f
        }

    // Per-element bias / gain (channel = 16*mt + r + rowOff)
    float bia[4][8], gai[4][8];
#pragma unroll
    for (int mt = 0; mt < 4; ++mt)
#pragma unroll
        for (int r = 0; r < 8; ++r) {
            const int ch = mt * 16 + r + rowOff;
            bia[mt][r] = b_out[ch];
            gai[mt][r] = g[ch];
        }

    const int tokensPerBlock = NTOK / Y_BLOCKS_PER_BATCH;  // 1024
    const int tokensPerWave  = tokensPerBlock / 4;         // 256
    const int tW0            = blk * tokensPerBlock + w * tokensPerWave;

    for (int tt = 0; tt < tokensPerWave; tt += 16) {
        const int t0 = tW0 + tt;
        v8f acc[4];
#pragma unroll
        for (int mt = 0; mt < 4; ++mt) acc[mt] = v8f_zero();

#pragma unroll
        for (int kg = 0; kg < 16; ++kg) {
            const float* bp = xb + (size_t)(kg * 4 + hi2) * NTOK + t0 + lanelo;
            v2f bop;
            bop[0] = bp[0];        // coalesced 64B per half-wave
            bop[1] = bp[NTOK];     // next channel, same token
#pragma unroll
            for (int mt = 0; mt < 4; ++mt)
                acc[mt] = wmma_f32(aop[kg][mt], bop, acc[mt]);
        }

        // Epilogue: +bias, LayerNorm over 64 channels (lanes L and L^16 share a voxel)
        float v[4][8];
        float s = 0.f, sq = 0.f;
#pragma unroll
        for (int mt = 0; mt < 4; ++mt)
#pragma unroll
            for (int r = 0; r < 8; ++r) {
                const float y = acc[mt][r] + bia[mt][r];
                v[mt][r] = y;
                s += y;
                sq += y * y;
            }
        s  += __shfl_xor(s, 16, 32);
        sq += __shfl_xor(sq, 16, 32);
        const float mean = s * (1.0f / 64.0f);
        const float var  = sq * (1.0f / 64.0f) - mean * mean;
        const float inv  = rsqrtf(var + 1e-5f);

#pragma unroll
        for (int mt = 0; mt < 4; ++mt)
#pragma unroll
            for (int r = 0; r < 8; ++r) {
                const int ch = mt * 16 + r + rowOff;
                ob[(size_t)ch * NTOK + t0 + lanelo] = (v[mt][r] - mean) * inv * gai[mt][r];
            }
    }
}

// ---------------------------------------------------------------- launch
extern "C" void kernel_launch(void* const* d_in, const int* in_sizes, int n_in,
                              void* d_out, int out_size, void* d_ws, size_t ws_size,
                              hipStream_t stream) {
    (void)in_sizes; (void)n_in; (void)out_size; (void)ws_size;
    const float* x    = (const float*)d_in[0];
    const float* wqkv = (const float*)d_in[1];
    const float* wout = (const float*)d_in[2];
    const float* bout = (const float*)d_in[3];
    const float* g    = (const float*)d_in[4];
    float* out        = (float*)d_out;
    float* ws         = (float*)d_ws;

    // d_ws layout (floats): G[2*4096] | A[2*4096] | T1[2*8192] | CT[2*4096] | M[2*8192]
    float* G  = ws;
    float* A  = ws + 8192;
    float* T1 = ws + 16384;
    float* CT = ws + 32768;
    float* M  = ws + 40960;   // total 57344 floats = 224 KB of scratch

    zero_kernel<<<32, 256, 0, stream>>>(G, 2 * CCH * CCH);
    gram_kernel<<<2 * G_BLOCKS_PER_BATCH, 128, 0, stream>>>(x, G);
    combine_kernel<<<2, 256, 0, stream>>>(wqkv, wout, G, T1, CT, M, A);
    attn_out_kernel<<<2 * Y_BLOCKS_PER_BATCH, 128, 0, stream>>>(x, A, bout, g, out);
}